// YearOperatorRouter_41575283425980
// MI455X (gfx1250) — compile-verified
//
#include <hip/hip_runtime.h>
#include <math.h>

// ---------------------------------------------------------------------------
// YearOperatorRouter on MI455X (gfx1250).
//
// Bandwidth-bound 3x3 depthwise stencil (AI ~2.4 FLOP/B, HBM floor ~6.5us at
// 23.3 TB/s). CDNA5 strategy:
//  * global_load_async_to_lds_b64 staging on the ASYNCcnt path. LDS pitch ==
//    global row pitch (1024B), so the hardware's shared INST_OFFSET (added to
//    both LDS and global addresses) lets all 17 DMAs/thread share two address
//    VGPRs and differ only by an immediate -> zero per-DMA VALU.
//  * column-pair-per-thread compute in ext_vector float2 so gx/gy/mag lower
//    to packed/dual f32 ops; B64 LDS loads and B64 global stores.
//  * 34-row strip (5.9% halo re-read), register-rotating row sweep.
// No WMMA by design: K=9 depthwise conv has no matrix-unit reuse to exploit.
// ---------------------------------------------------------------------------

typedef float v2f __attribute__((ext_vector_type(2)));

#define IMG     256
#define CHANS   9
#define BATCH   32
#define STRIP   32                  // output rows per block (8 strips/plane)
#define NSTRIPS (IMG / STRIP)
#define LROWS   (STRIP + 2)         // staged rows incl. top/bottom halo
// LDS tile: LROWS x IMG floats, pitch 1024B == global pitch. 34 KB.

#define ASYNC_B64(lds, goff, base, imm)                                     \
    asm volatile("global_load_async_to_lds_b64 %0, %1, %2 offset:%3"        \
                 :: "v"(lds), "v"(goff), "s"(base), "i"(imm) : "memory")

__global__ __launch_bounds__(256) void year_router_edge_kernel(
    const float* __restrict__ x,
    const float* __restrict__ alpha,
    const float* __restrict__ scale,
    const int*   __restrict__ year,
    float*       __restrict__ out)
{
    __shared__ float tile[LROWS * IMG];          // 34816 B

    const int tid   = threadIdx.x;               // 0..255
    const int strip = blockIdx.x;                // 0..7
    const int plane = blockIdx.y;                // 0..287 (b*CHANS + c)
    const int r0    = strip * STRIP;

    const float* pbase = x + (size_t)plane * (IMG * IMG);

    // ---- Stage 34 rows (global rows r0-1 .. r0+32) as 4352 float2 elements,
    // 17 b64 DMAs per thread. Element e = tid + 256*k -> LDS byte e*8,
    // global byte (r0-1)*1024 + e*8. Shared stride => immediate offsets only.
    const unsigned lds0 = (unsigned)(size_t)&tile[0] + (unsigned)(tid * 8);
    const int      g0   = (r0 - 1) * (IMG * 4) + tid * 8;

    // k = 0 covers LDS rows 0 (tid<128) and 1 (tid>=128); row 0 is the top
    // halo -> zero-fill for the first strip instead of loading row -1.
    if (strip == 0 && tid < 128) {
        *(v2f*)&tile[tid * 2] = (v2f)0.0f;
    } else {
        ASYNC_B64(lds0, g0, pbase, 0);
    }
#pragma unroll
    for (int k = 1; k < 16; ++k) {               // rows 2..31: always valid
        ASYNC_B64(lds0, g0, pbase, k * 2048);
    }
    // k = 16 covers LDS rows 32 (tid<128) and 33 (tid>=128); row 33 is the
    // bottom halo -> zero-fill for the last strip instead of loading row 256.
    if (strip == NSTRIPS - 1 && tid >= 128) {
        *(v2f*)&tile[8192 + tid * 2] = (v2f)0.0f;
    } else {
        ASYNC_B64(lds0, g0, pbase, 16 * 2048);
    }

    asm volatile("s_wait_asynccnt 0" ::: "memory");   // this wave's DMAs done
    __syncthreads();                                   // visible across waves

    // ---- Per-sample operator parameters (uniform per block).
    // kx = [[-p,0,p],[-q,0,q],[-p,0,p]], ky = kx^T (XLA conv = cross-corr.)
    const int yr = year[plane / CHANS];
    float p, q, eps, post;
    if (yr == 2019) {            // Sobel
        p = 1.0f;  q = 2.0f;  eps = 1e-8f; post = 1.0f;
    } else if (yr == 2020) {     // Scharr
        p = 3.0f;  q = 10.0f; eps = 1e-8f; post = 1.0f;
    } else {                     // learnable sigmoid-mix, max-abs normalized
        const float a = 1.0f / (1.0f + expf(-alpha[0]));
        const float m = 10.0f - 8.0f * a;      // center coeff = max |mixed|
        p = (3.0f - 2.0f * a) / m;
        q = 1.0f;
        eps  = 1e-6f;
        post = 1.0f / (1.0f + expf(-scale[0]));
    }
    const v2f p2 = {p, p}, q2 = {q, q}, eps2 = {eps, eps};

    // ---- Compute mapping: 128 column pairs x 2 row groups of 16 rows.
    const int cp  = tid & 127;                   // columns 2cp, 2cp+1
    const int rg  = tid >> 7;                    // row group 0..1
    const int li0 = rg * 16;                     // LDS row window li0..li0+17
    const bool eL = (cp == 0);                   // image left edge
    const bool eR = (cp == 127);                 // image right edge

    float* obase = out + (size_t)plane * (IMG * IMG)
                       + (size_t)(r0 + li0) * IMG + cp * 2;

    // Row terms: l | (c.x c.y) | r ; packed L=(l,c.x), R=(c.y,r), C=c.
    auto load_row = [&](int lrow, v2f& L, v2f& R, v2f& C) {
        const int bi = lrow * IMG + cp * 2;
        const v2f c  = *(const v2f*)&tile[bi];
        const float l = eL ? 0.0f : tile[bi - 1];   // OOB-safe: selected away
        const float r = eR ? 0.0f : tile[bi + 2];
        L = (v2f){l, c.x};
        R = (v2f){c.y, r};
        C = c;
    };

    v2f LA, RA, CA, LB, RB, CB;
    load_row(li0 + 0, LA, RA, CA);
    load_row(li0 + 1, LB, RB, CB);
    v2f dA = RA - LA;                            // horizontal differences
    v2f dB = RB - LB;

#pragma unroll
    for (int i = 0; i < 16; ++i) {
        v2f LC, RC, CC;
        load_row(li0 + i + 2, LC, RC, CC);
        const v2f dC = RC - LC;

        // gx = p*(dA + dC) + q*dB
        const v2f gx = __builtin_elementwise_fma(q2, dB, p2 * (dA + dC));
        // gy = p*((LC-LA) + (RC-RA)) + q*(CC-CA)
        const v2f gy = __builtin_elementwise_fma(q2, CC - CA,
                                                 p2 * ((LC - LA) + (RC - RA)));
        const v2f m  = __builtin_elementwise_fma(
                           gx, gx, __builtin_elementwise_fma(gy, gy, eps2));
        v2f o;
        o.x = sqrtf(m.x);
        o.y = sqrtf(m.y);
        o = o * post;

        *(v2f*)&obase[i * IMG] = o;

        LA = LB; RA = RB; CA = CB; dA = dB;
        LB = LC; RB = RC; CB = CC; dB = dC;
    }
}

extern "C" void kernel_launch(void* const* d_in, const int* in_sizes, int n_in,
                              void* d_out, int out_size, void* d_ws, size_t ws_size,
                              hipStream_t stream)
{
    (void)in_sizes; (void)n_in; (void)out_size; (void)d_ws; (void)ws_size;

    const float* x     = (const float*)d_in[0];   // [32,9,256,256] f32
    const float* alpha = (const float*)d_in[1];   // scalar f32
    const float* scale = (const float*)d_in[2];   // [1] f32
    const int*   year  = (const int*)  d_in[3];   // [32] i32
    float*       out   = (float*)d_out;           // [32,9,256,256] f32

    dim3 grid(NSTRIPS, BATCH * CHANS);            // (8, 288)
    dim3 block(256);                              // 8 waves (wave32)
    year_router_edge_kernel<<<grid, block, 0, stream>>>(x, alpha, scale, year, out);
}